// GraphNet_54666343743722
// MI455X (gfx1250) — compile-verified
//
#include <hip/hip_runtime.h>

typedef float v2f __attribute__((ext_vector_type(2)));
typedef float v8f __attribute__((ext_vector_type(8)));

#define C 64   // feature width of every hidden layer (and input)

// ---------------------------------------------------------------- utilities
__global__ __launch_bounds__(256) void k_zero(float* __restrict__ p, int n) {
  int i = blockIdx.x * 256 + threadIdx.x;
  if (i < n) p[i] = 0.0f;
}

__global__ __launch_bounds__(256) void k_degcount(const int* __restrict__ row,
                                                  float* __restrict__ deg,
                                                  int n_edges) {
  int e = blockIdx.x * 256 + threadIdx.x;
  if (e < n_edges) atomicAdd(&deg[row[e]], 1.0f);
}

__global__ __launch_bounds__(256) void k_invdeg(float* __restrict__ deg, int n) {
  int i = blockIdx.x * 256 + threadIdx.x;
  if (i < n) deg[i] = 1.0f / fmaxf(deg[i], 1.0f);
}

// ------------------------------------------------- edge scatter (sum agg)
// 16 threads per edge, 4 channels each (float4 gather + 4 f32 atomics).
// Whole working set is L2-resident on MI455X (192MB L2), so these are
// L2-atomic-rate bound, not HBM bound.
__global__ __launch_bounds__(256) void k_scatter(const float* __restrict__ src,
                                                 const int* __restrict__ row,
                                                 const int* __restrict__ col,
                                                 float* __restrict__ agg,
                                                 int n_edges) {
  unsigned t = blockIdx.x * 256u + threadIdx.x;
  unsigned e = t >> 4;
  if (e >= (unsigned)n_edges) return;
  int c = (int)(t & 15u) << 2;
  int r = row[e];
  int s = col[e];
  float4 v = *(const float4*)(src + s * C + c);
  float* d = agg + r * C + c;
  atomicAdd(d + 0, v.x);
  atomicAdd(d + 1, v.y);
  atomicAdd(d + 2, v.z);
  atomicAdd(d + 3, v.w);
}

// ------------------------------------------------- WMMA GEMM + bias + ReLU
// out[n, o] = act( sum_k X[n,k]*W[o,k] + sum_k (AGG[n,k]*invdeg[n])*W[o,64+k] + B[o] )
// One wave32 computes a 16-node x 16-outch tile with V_WMMA_F32_16X16X4_F32
// (exact f32 path). K=128 handled as two 64-wide phases (x-half, agg-half),
// 16 WMMA steps of K=4 each.
//
// f32 16x4 A layout (ISA 7.12.2): lane L -> row M = L%16,
//   VGPR0 = A[M, kk + (L>=16?2:0) + 0], VGPR1 = A[M, kk + (L>=16?2:0) + 1]
// B (4x16) is the symmetric layout over columns N = L%16.
// C/D: VGPR i -> (M = i + (L>=16?8:0), N = L%16).
// Requires n_nodes % 16 == 0 (100000 = 6250*16).
__global__ __launch_bounds__(256) void k_sage_gemm(const float* __restrict__ X,
                                                   const float* __restrict__ AGG,
                                                   const float* __restrict__ INVDEG,
                                                   const float* __restrict__ W,   // [64,128]
                                                   const float* __restrict__ B,   // [64]
                                                   float* __restrict__ OUT,       // [N,64]
                                                   int n_nodes, int do_relu) {
  int wave = (blockIdx.x * 256 + threadIdx.x) >> 5;
  int lane = threadIdx.x & 31;
  int tiles_m = n_nodes >> 4;
  int tile_m = wave >> 2;        // 4 col-tiles (of 16 out channels) per row-tile
  int o_tile = wave & 3;
  if (tile_m >= tiles_m) return; // wave-uniform

  int l16   = lane & 15;
  int khalf = (lane >> 4) << 1;          // 0 for lanes 0-15, 2 for lanes 16-31
  int m     = (tile_m << 4) + l16;       // node row this lane feeds into A
  int o_base = o_tile << 4;

  const float* xrow = X + m * C + khalf;
  const float* grow = AGG + m * C + khalf;
  const float* wrowL = W + (o_base + l16) * (2 * C) + khalf;      // W_left
  const float* wrowR = wrowL + C;                                 // W_right

  v8f acc = {};

  // phase 1: self features  x . W_left
#pragma unroll
  for (int kk = 0; kk < C; kk += 4) {
    v2f a = *(const v2f*)(xrow + kk);
    v2f b = *(const v2f*)(wrowL + kk);
    acc = __builtin_amdgcn_wmma_f32_16x16x4_f32(false, a, false, b,
                                                (short)0, acc, false, false);
  }

  // phase 2: mean-aggregated neighbors  (agg*invdeg) . W_right
  float s = INVDEG[m];
#pragma unroll
  for (int kk = 0; kk < C; kk += 4) {
    v2f a = *(const v2f*)(grow + kk);
    a.x *= s;
    a.y *= s;
    v2f b = *(const v2f*)(wrowR + kk);
    acc = __builtin_amdgcn_wmma_f32_16x16x4_f32(false, a, false, b,
                                                (short)0, acc, false, false);
  }

  // epilogue: bias + ReLU + store (D layout: VGPR i -> M = i + (lane>=16?8:0))
  float bias = B[o_base + l16];
  int mbase = (tile_m << 4) + ((lane >> 4) << 3);
  int n_out = o_base + l16;
#pragma unroll
  for (int i = 0; i < 8; ++i) {
    float v = acc[i] + bias;
    if (do_relu) v = fmaxf(v, 0.0f);
    OUT[(mbase + i) * C + n_out] = v;
  }
}

// ------------------------------------------------- final layer (OUT_CH = 1)
__global__ __launch_bounds__(256) void k_sage_out(const float* __restrict__ H,
                                                  const float* __restrict__ AGG,
                                                  const float* __restrict__ INVDEG,
                                                  const float* __restrict__ W3,  // [1,128]
                                                  const float* __restrict__ B3,  // [1]
                                                  float* __restrict__ OUT,       // [N]
                                                  int n_nodes) {
  int i = blockIdx.x * 256 + threadIdx.x;
  if (i >= n_nodes) return;
  float s = INVDEG[i];
  const float* h = H + i * C;
  const float* g = AGG + i * C;
  float acc = 0.0f;
#pragma unroll 8
  for (int k = 0; k < C; ++k) acc += h[k] * W3[k];
#pragma unroll 8
  for (int k = 0; k < C; ++k) acc += g[k] * s * W3[C + k];
  OUT[i] = acc + B3[0];
}

// ---------------------------------------------------------------- launcher
extern "C" void kernel_launch(void* const* d_in, const int* in_sizes, int n_in,
                              void* d_out, int out_size, void* d_ws, size_t ws_size,
                              hipStream_t stream) {
  const float* x  = (const float*)d_in[0];
  const int* eidx = (const int*)d_in[1];
  const float* W1 = (const float*)d_in[2];
  const float* b1 = (const float*)d_in[3];
  const float* W2 = (const float*)d_in[4];
  const float* b2 = (const float*)d_in[5];
  const float* W3 = (const float*)d_in[6];
  const float* b3 = (const float*)d_in[7];

  int n_nodes = in_sizes[0] / C;  // 100000
  int n_edges = in_sizes[1] / 2;  // 1600000
  const int* row = eidx;            // edge_index[0]
  const int* col = eidx + n_edges;  // edge_index[1]

  // workspace layout (floats): invdeg | agg | h1 | h2
  float* ws = (float*)d_ws;
  int nf = n_nodes * C;
  int pad = (n_nodes + 255) & ~255;
  float* invdeg = ws;
  float* agg = ws + pad;
  float* h1  = agg + nf;
  float* h2  = h1 + nf;

  int zb_n   = (n_nodes + 255) / 256;
  int zb_nf  = (nf + 255) / 256;
  int eb     = (n_edges + 255) / 256;
  int sb     = (int)(((long long)n_edges * 16 + 255) / 256);
  int waves  = (n_nodes / 16) * 4;
  int gb     = (waves * 32 + 255) / 256;

  // degrees (shared by all three layers)
  k_zero<<<zb_n, 256, 0, stream>>>(invdeg, n_nodes);
  k_degcount<<<eb, 256, 0, stream>>>(row, invdeg, n_edges);
  k_invdeg<<<zb_n, 256, 0, stream>>>(invdeg, n_nodes);

  // layer 1: x -> h1
  k_zero<<<zb_nf, 256, 0, stream>>>(agg, nf);
  k_scatter<<<sb, 256, 0, stream>>>(x, row, col, agg, n_edges);
  k_sage_gemm<<<gb, 256, 0, stream>>>(x, agg, invdeg, W1, b1, h1, n_nodes, 1);

  // layer 2: h1 -> h2
  k_zero<<<zb_nf, 256, 0, stream>>>(agg, nf);
  k_scatter<<<sb, 256, 0, stream>>>(h1, row, col, agg, n_edges);
  k_sage_gemm<<<gb, 256, 0, stream>>>(h1, agg, invdeg, W2, b2, h2, n_nodes, 1);

  // layer 3: h2 -> out [N,1]
  k_zero<<<zb_nf, 256, 0, stream>>>(agg, nf);
  k_scatter<<<sb, 256, 0, stream>>>(h2, row, col, agg, n_edges);
  k_sage_out<<<zb_n, 256, 0, stream>>>(h2, agg, invdeg, W3, b3, (float*)d_out,
                                       n_nodes);
}